// GRE_48558900248706
// MI455X (gfx1250) — compile-verified
//
#include <hip/hip_runtime.h>

#define T_SIZE  8192
#define KMIX    10
#define D_MODEL 256
#define BATCH   32

typedef __attribute__((ext_vector_type(2))) float v2f;
typedef __attribute__((ext_vector_type(4))) float v4f;
typedef __attribute__((ext_vector_type(8))) float v8f;

// ---------------------------------------------------------------------------
// Kernel 1: pos_enc = softmax(log_p) @ embedding  via V_WMMA_F32_16X16X4_F32
// Grid: (512 M-tiles, 16 N-tiles), one wave32 per 16x16 output tile.
// K = 10 padded to 12 -> 3 WMMA steps of K=4.
// All operand selection is done via lane-half ternaries / address math:
// no dynamic register indexing, no divergent branches around loads.
// ---------------------------------------------------------------------------
__global__ __launch_bounds__(32) void GRE_pe_wmma_kernel(
    const float* __restrict__ emb,    // (10, 256)
    const float* __restrict__ mu,     // (10)
    const float* __restrict__ sigma,  // (10)
    float* __restrict__ pe)           // (8192, 256)
{
  const int lane = threadIdx.x & 31;
  const int hi   = lane >> 4;   // 0: lanes 0-15, 1: lanes 16-31
  const int l16  = lane & 15;
  const int tm   = blockIdx.x;  // 0..511  (rows 16*tm .. 16*tm+15)
  const int tn   = blockIdx.y;  // 0..15   (cols 16*tn .. 16*tn+15)

  // --- per-row softmax weights (row = tm*16 + l16; both halves same row) ---
  const float t = (float)(tm * 16 + l16);
  float lp[KMIX];
  float mx = -3.0e38f;
#pragma unroll
  for (int k = 0; k < KMIX; ++k) {
    const float s = sigma[k];                 // uniform -> s_load
    const float a = t - mu[k];
    const float l = -(a * a) / (2.0f * s + 1e-8f)
                    - 0.5f * __logf(fabsf(s) + 1e-8f);
    lp[k] = l;
    mx = fmaxf(mx, l);
  }
  float sum = 0.0f;
#pragma unroll
  for (int k = 0; k < KMIX; ++k) {
    lp[k] = __expf(lp[k] - mx);
    sum += lp[k];
  }
  const float inv = 1.0f / sum;
#pragma unroll
  for (int k = 0; k < KMIX; ++k) lp[k] *= inv;   // lp[] = softmax weights

  const int n = tn * 16 + l16;          // B/C column handled by this lane
  const float* ecol = emb + n;          // column n of embedding
  const int hoff = hi * 2 * D_MODEL;    // lane-half K-pair address offset

  // --- B operand loads: emb[k][n], k = 4s + 2*hi (+1); unconditional ----
  v2f b0, b1, b2;
  b0.x = ecol[hoff];                         // k = 0 | 2
  b0.y = ecol[hoff + D_MODEL];               // k = 1 | 3
  b1.x = ecol[hoff + 4 * D_MODEL];           // k = 4 | 6
  b1.y = ecol[hoff + 5 * D_MODEL];           // k = 5 | 7
  const float e8 = ecol[8 * D_MODEL];        // k = 8
  const float e9 = ecol[9 * D_MODEL];        // k = 9
  b2.x = hi ? 0.0f : e8;                     // k = 8 | pad(10)
  b2.y = hi ? 0.0f : e9;                     // k = 9 | pad(11)

  // --- A operands: lane-half selects the K pair (plain cndmask selects) --
  v2f a0, a1, a2;
  a0.x = hi ? lp[2] : lp[0];
  a0.y = hi ? lp[3] : lp[1];
  a1.x = hi ? lp[6] : lp[4];
  a1.y = hi ? lp[7] : lp[5];
  a2.x = hi ? 0.0f : lp[8];
  a2.y = hi ? 0.0f : lp[9];

  // --- 3x V_WMMA_F32_16X16X4_F32, accumulate 16x16 f32 tile -------------
  v8f c = {};
  c = __builtin_amdgcn_wmma_f32_16x16x4_f32(false, a0, false, b0, (short)0, c,
                                            false, false);
  c = __builtin_amdgcn_wmma_f32_16x16x4_f32(false, a1, false, b1, (short)0, c,
                                            false, false);
  c = __builtin_amdgcn_wmma_f32_16x16x4_f32(false, a2, false, b2, (short)0, c,
                                            false, false);

  // --- store C: VGPR v -> row (tm*16 + hi*8 + v), col n ------------------
  float* prow = pe + (tm * 16 + hi * 8) * D_MODEL + n;
#pragma unroll
  for (int v = 0; v < 8; ++v) {
    prow[v * D_MODEL] = c[v];
  }
}

// ---------------------------------------------------------------------------
// Kernel 2: out[b,t,d] = x[b,t,d] + pe[t,d]  — pure 16B-coalesced stream.
// x/out are non-temporal (TH=NT) so the 537 MB stream doesn't evict the
// 8 MB pe table from the 192 MB L2 across its 32x batch reuse.
// ---------------------------------------------------------------------------
__global__ __launch_bounds__(256) void GRE_add_pe_kernel(
    const v4f* __restrict__ x,
    const v4f* __restrict__ pe,
    v4f* __restrict__ out)
{
  const int idx = blockIdx.x * 256 + threadIdx.x;          // < 16,777,216
  const int p   = idx & (T_SIZE * D_MODEL / 4 - 1);        // 2^19 float4s
  const v4f a = __builtin_nontemporal_load(&x[idx]);
  const v4f b = pe[p];                                     // L2-resident
  const v4f r = a + b;
  __builtin_nontemporal_store(r, &out[idx]);
}

// ---------------------------------------------------------------------------
extern "C" void kernel_launch(void* const* d_in, const int* in_sizes, int n_in,
                              void* d_out, int out_size, void* d_ws, size_t ws_size,
                              hipStream_t stream) {
  const float* x     = (const float*)d_in[0];  // (32, 8192, 256)
  const float* emb   = (const float*)d_in[1];  // (10, 256)
  const float* mu    = (const float*)d_in[2];  // (10)
  const float* sigma = (const float*)d_in[3];  // (10)
  float* out = (float*)d_out;
  float* pe  = (float*)d_ws;                   // 8192*256*4 = 8 MB scratch

  // 1) pos_enc table via WMMA (tiny: 8 MB of output, 8192 wave32 tiles)
  GRE_pe_wmma_kernel<<<dim3(T_SIZE / 16, D_MODEL / 16), 32, 0, stream>>>(
      emb, mu, sigma, pe);

  // 2) bandwidth-bound broadcast add: 537 MB stream -> ~23 us roofline
  const int total4 = BATCH * T_SIZE * D_MODEL / 4;  // 16,777,216
  GRE_add_pe_kernel<<<total4 / 256, 256, 0, stream>>>(
      (const v4f*)x, (const v4f*)pe, (v4f*)out);
}